// GCN_41489384079831
// MI455X (gfx1250) — compile-verified
//
#include <hip/hip_runtime.h>
#include <stdint.h>
#include <math.h>

typedef __attribute__((ext_vector_type(2))) float v2f;
typedef __attribute__((ext_vector_type(8))) float v8f;

#define THREADS 256

// ---------------- init: zero accumulators, deg = 1 (self loop) ----------------
__global__ void init_kernel(float* __restrict__ agg1, float* __restrict__ outb,
                            float* __restrict__ deg, int n) {
  int t = blockIdx.x * blockDim.x + threadIdx.x;
  if (t < n * 32) agg1[t] = 0.0f;
  if (t < n * 16) outb[t] = 0.0f;
  if (t < n)      deg[t]  = 1.0f;
}

// ---------------- degree accumulation over dst ----------------
__global__ void deg_kernel(const long long* __restrict__ dst, float* __restrict__ deg, int e) {
  int t = blockIdx.x * blockDim.x + threadIdx.x;
  if (t < e) atomicAdd(&deg[dst[t]], 1.0f);
}

__global__ void rsqrt_kernel(float* __restrict__ deg, int n) {
  int t = blockIdx.x * blockDim.x + threadIdx.x;
  if (t < n) deg[t] = rsqrtf(deg[t]);
}

// ---------------- GEMM1: H[n,32] = X[n,64] @ W[64,32] via f32 WMMA ----------------
// One wave per 16-row stripe; two 16-wide N tiles share the A loads.
__global__ void gemm1_wmma(const float* __restrict__ X, const float* __restrict__ W,
                           float* __restrict__ H, int n) {
  int wave = (int)((blockIdx.x * blockDim.x + threadIdx.x) >> 5);
  int lane = threadIdx.x & 31;
  int m0 = wave << 4;
  if (m0 >= n) return;                 // wave-uniform: EXEC stays all-ones
  int half = lane >> 4;                // 0: lanes 0-15, 1: lanes 16-31
  int l15  = lane & 15;
  int arow = m0 + l15; if (arow >= n) arow = n - 1;
  const float* xrow = X + (size_t)arow * 64;

  v8f c0 = {}, c1 = {};
#pragma unroll
  for (int k = 0; k < 64; k += 4) {
    int ka = k + (half << 1);          // A: VGPR pair holds K=ka,ka+1 in this lane-half
    v2f a;  a.x  = xrow[ka];                 a.y  = xrow[ka + 1];
    v2f b0; b0.x = W[ka * 32 + l15];         b0.y = W[(ka + 1) * 32 + l15];
    v2f b1; b1.x = W[ka * 32 + 16 + l15];    b1.y = W[(ka + 1) * 32 + 16 + l15];
    c0 = __builtin_amdgcn_wmma_f32_16x16x4_f32(false, a, false, b0, (short)0, c0, false, false);
    c1 = __builtin_amdgcn_wmma_f32_16x16x4_f32(false, a, false, b1, (short)0, c1, false, false);
  }
#pragma unroll
  for (int r = 0; r < 8; ++r) {
    int row = m0 + r + (half << 3);    // C/D: VGPR r -> M=r (lanes 0-15) / M=8+r (16-31)
    if (row < n) {
      H[(size_t)row * 32 + l15]      = c0[r];
      H[(size_t)row * 32 + 16 + l15] = c1[r];
    }
  }
}

// ---------------- GEMM2: H[n,16] = Y[n,32] @ W[32,16] via f32 WMMA ----------------
__global__ void gemm2_wmma(const float* __restrict__ Y, const float* __restrict__ W,
                           float* __restrict__ H, int n) {
  int wave = (int)((blockIdx.x * blockDim.x + threadIdx.x) >> 5);
  int lane = threadIdx.x & 31;
  int m0 = wave << 4;
  if (m0 >= n) return;
  int half = lane >> 4;
  int l15  = lane & 15;
  int arow = m0 + l15; if (arow >= n) arow = n - 1;
  const float* yrow = Y + (size_t)arow * 32;

  v8f c = {};
#pragma unroll
  for (int k = 0; k < 32; k += 4) {
    int ka = k + (half << 1);
    v2f a; a.x = yrow[ka];            a.y = yrow[ka + 1];
    v2f b; b.x = W[ka * 16 + l15];    b.y = W[(ka + 1) * 16 + l15];
    c = __builtin_amdgcn_wmma_f32_16x16x4_f32(false, a, false, b, (short)0, c, false, false);
  }
#pragma unroll
  for (int r = 0; r < 8; ++r) {
    int row = m0 + r + (half << 3);
    if (row < n) H[(size_t)row * 16 + l15] = c[r];
  }
}

// ---------------- edge scatter: acc[dst] += H[src] * dinv[src]*dinv[dst] ----------------
// (4<<LOG_G) channels per edge; (1<<LOG_G) threads per edge, float4 per thread.
template <int LOG_G>
__global__ void scatter_kernel(const float* __restrict__ H,
                               const long long* __restrict__ src,
                               const long long* __restrict__ dst,
                               const float* __restrict__ dinv,
                               float* __restrict__ acc, int e) {
  long long t = (long long)blockIdx.x * blockDim.x + threadIdx.x;
  long long eid = t >> LOG_G;
  if (eid >= e) return;
  const int C   = 4 << LOG_G;
  const int sub = (int)t & ((1 << LOG_G) - 1);
  long long s = src[eid], d = dst[eid];
  float w = dinv[s] * dinv[d];
  const float4 h = *(const float4*)(H + (size_t)s * C + sub * 4);
  float* o = acc + (size_t)d * C + sub * 4;
  atomicAdd(o + 0, h.x * w);
  atomicAdd(o + 1, h.y * w);
  atomicAdd(o + 2, h.z * w);
  atomicAdd(o + 3, h.w * w);
}

// ---------------- fused self-loop + bias + ReLU (in place into H1) ----------------
__global__ void post1_kernel(float* __restrict__ H1, const float* __restrict__ AGG,
                             const float* __restrict__ dinv, const float* __restrict__ b, int n) {
  int t = blockIdx.x * blockDim.x + threadIdx.x;
  if (t >= n * 32) return;
  int i = t >> 5, j = t & 31;
  float dv = dinv[i]; dv *= dv;                 // self-loop norm = dinv[i]^2
  float v = AGG[t] + H1[t] * dv + b[j];
  H1[t] = fmaxf(v, 0.0f);
}

// ---------------- fused self-loop + bias + log_softmax (in place into OUT) ----------------
__global__ void post2_kernel(float* __restrict__ OUT, const float* __restrict__ H2,
                             const float* __restrict__ dinv, const float* __restrict__ b, int n) {
  int i = blockIdx.x * blockDim.x + threadIdx.x;
  if (i >= n) return;
  float dv = dinv[i]; dv *= dv;
  float v[16];
  float m = -INFINITY;
#pragma unroll
  for (int j = 0; j < 16; ++j) {
    v[j] = OUT[(size_t)i * 16 + j] + H2[(size_t)i * 16 + j] * dv + b[j];
    m = fmaxf(m, v[j]);
  }
  float s = 0.0f;
#pragma unroll
  for (int j = 0; j < 16; ++j) s += expf(v[j] - m);
  float lse = m + logf(s);
#pragma unroll
  for (int j = 0; j < 16; ++j) OUT[(size_t)i * 16 + j] = v[j] - lse;
}

extern "C" void kernel_launch(void* const* d_in, const int* in_sizes, int n_in,
                              void* d_out, int out_size, void* d_ws, size_t ws_size,
                              hipStream_t stream) {
  const float*     X  = (const float*)d_in[0];
  const long long* EI = (const long long*)d_in[1];   // int64 edge_index [2, E]
  const float*     W1 = (const float*)d_in[2];
  const float*     b1 = (const float*)d_in[3];
  const float*     W2 = (const float*)d_in[4];
  const float*     b2 = (const float*)d_in[5];
  float* out = (float*)d_out;

  const int n = in_sizes[0] / 64;
  const int e = in_sizes[1] / 2;
  const long long* src = EI;
  const long long* dst = EI + e;

  // Workspace layout (256B-aligned slabs): dinv[n] | h1[n*32] | agg1[n*32] | h2[n*16]
  const size_t na = ((size_t)n + 63) & ~(size_t)63;
  float* ws   = (float*)d_ws;
  float* deg  = ws;                       // becomes dinv in place
  float* h1   = ws + na;
  float* agg1 = h1 + na * 32;
  float* h2   = agg1 + na * 32;

  const int n32 = n * 32;
  const dim3 blk(THREADS);
  const int gb_n32 = (n32 + THREADS - 1) / THREADS;
  const int gb_n   = (n + THREADS - 1) / THREADS;
  const int gb_e   = (e + THREADS - 1) / THREADS;
  const int waves  = (n + 15) / 16;
  const int gb_mm  = (waves + 7) / 8;     // 8 waves per 256-thread block

  init_kernel<<<gb_n32, blk, 0, stream>>>(agg1, out, deg, n);
  deg_kernel<<<gb_e, blk, 0, stream>>>(dst, deg, e);
  rsqrt_kernel<<<gb_n, blk, 0, stream>>>(deg, n);

  gemm1_wmma<<<gb_mm, blk, 0, stream>>>(X, W1, h1, n);
  scatter_kernel<3><<<(int)(((long long)e * 8 + THREADS - 1) / THREADS), blk, 0, stream>>>(
      h1, src, dst, deg, agg1, e);
  post1_kernel<<<gb_n32, blk, 0, stream>>>(h1, agg1, deg, b1, n);

  gemm2_wmma<<<gb_mm, blk, 0, stream>>>(h1, W2, h2, n);
  scatter_kernel<2><<<(int)(((long long)e * 4 + THREADS - 1) / THREADS), blk, 0, stream>>>(
      h2, src, dst, deg, out, e);
  post2_kernel<<<gb_n, blk, 0, stream>>>(out, h2, deg, b2, n);
}